// DynamicConvBlock_84765474554354
// MI455X (gfx1250) — compile-verified
//
#include <hip/hip_runtime.h>

#define C_CH 384
#define N_IMG 16
#define HW 56
#define KK 13
#define PW 44   // patch width: 32 outputs + 12 halo
#define PH 28   // patch height: 16 outputs + 12 halo
#define WROW 44 // padded Toeplitz weight row length
#define TPAD 15 // maps t in [-15..28] -> [0..43]

typedef __attribute__((ext_vector_type(2))) float v2f;
typedef __attribute__((ext_vector_type(8))) float v8f;

// ---------------- Phase 1: fold 7 BN'd branches into one 13x13 DW kernel ----
__device__ __forceinline__ float branch_tap(const float* __restrict__ w, int c,
                                            int k, int r, float s, int dy, int dx) {
    if ((dy % r) != 0 || (dx % r) != 0) return 0.0f;
    int hk = (k - 1) >> 1;
    int ky = dy / r + hk;
    int kx = dx / r + hk;
    if (ky < 0 || ky >= k || kx < 0 || kx >= k) return 0.0f;
    return s * w[(c * k + ky) * k + kx];
}

__global__ void merge_weights_kernel(const float* __restrict__ w_lk,
                                     const float* __restrict__ w_b0,
                                     const float* __restrict__ w_b1,
                                     const float* __restrict__ w_b2,
                                     const float* __restrict__ w_b3,
                                     const float* __restrict__ w_b4,
                                     const float* __restrict__ w_b5,
                                     const float* __restrict__ gamma,
                                     const float* __restrict__ beta,
                                     const float* __restrict__ mean,
                                     const float* __restrict__ var,
                                     float* __restrict__ Wm,
                                     float* __restrict__ bias) {
    const int c = blockIdx.x;
    const int p = threadIdx.x;
    float s[7];
    #pragma unroll
    for (int j = 0; j < 7; ++j)
        s[j] = gamma[j * C_CH + c] * rsqrtf(var[j * C_CH + c] + 1e-5f);
    if (p < 169) {
        int ty = p / 13, tx = p % 13;
        int dy = ty - 6, dx = tx - 6;
        float acc = s[0] * w_lk[c * 169 + p];
        acc += branch_tap(w_b0, c, 5, 1, s[1], dy, dx);
        acc += branch_tap(w_b1, c, 7, 1, s[2], dy, dx);
        acc += branch_tap(w_b2, c, 7, 2, s[3], dy, dx);
        acc += branch_tap(w_b3, c, 3, 3, s[4], dy, dx);
        acc += branch_tap(w_b4, c, 3, 4, s[5], dy, dx);
        acc += branch_tap(w_b5, c, 3, 5, s[6], dy, dx);
        Wm[c * 169 + p] = acc;
    }
    if (p == 0) {
        float b = 0.0f;
        #pragma unroll
        for (int j = 0; j < 7; ++j)
            b += beta[j * C_CH + c] - mean[j * C_CH + c] * s[j];
        bias[c] = b;
    }
}

// ---------------- Phase 2: 13x13 depthwise conv via Toeplitz f32 WMMA -------
// One wave computes a 16(y) x 32(x) output tile for one (n, c).
// For each kernel row ky and K-chunk, both 16x16 halves share the same
// Toeplitz B (it depends only on xi - x); A is the patch row, shifted by 16
// columns for the right half. 2 WMMAs per B pair, 182 WMMAs per wave,
// fully unrolled so accumulators stay pinned in VGPRs (no copies/branches).
__global__ void __launch_bounds__(32)
dwconv13_wmma_kernel(const float* __restrict__ xin,
                     const float* __restrict__ Wm,
                     const float* __restrict__ bias,
                     float* __restrict__ out) {
    __shared__ float patch[PH * PW];      // 28x44 input patch, zero-padded
    __shared__ float wpad[KK * WROW];     // per-ky weight row, zero-padded band

    const int lid  = threadIdx.x;         // 0..31
    const int tile = blockIdx.x;          // 0..7: (ytile 0..3) x (xpair 0..1)
    const int c    = blockIdx.y;
    const int n    = blockIdx.z;
    const int y0   = (tile >> 1) * 16;
    const int x0   = (tile & 1) * 32;

    // zero padded weight rows, then fill the 13 valid taps per row
    for (int i = lid; i < KK * WROW; i += 32) wpad[i] = 0.0f;
    __syncthreads();
    const float* wc = Wm + c * 169;
    for (int i = lid; i < 169; i += 32) {
        int ky = i / 13, kx = i % 13;
        wpad[ky * WROW + TPAD + kx] = wc[i];
    }

    // stage 28x44 input patch with zero fill at image borders
    const float* xc = xin + ((size_t)(n * C_CH + c)) * (HW * HW);
    for (int i = lid; i < PH * PW; i += 32) {
        int py = i / PW, px = i % PW;
        int gy = y0 + py - 6, gx = x0 + px - 6;
        float v = 0.0f;
        if (gy >= 0 && gy < HW && gx >= 0 && gx < HW)
            v = xc[gy * HW + gx];
        patch[i] = v;
    }
    __syncthreads();

    const int mn = lid & 15;   // A: M row; B/D: N column
    const int h  = lid >> 4;   // lane half selects K pair {2h, 2h+1}

    const float* prow0 = patch + (mn)*PW + 2 * h;           // A base (ky = 0)
    const float* wrow0 = wpad + TPAD + 2 * h - mn;          // B base (ky = 0)

    v8f accL = {};
    v8f accR = {};
    #pragma unroll
    for (int ky = 0; ky < KK; ++ky) {
        const float* prow = prow0 + ky * PW;
        const float* wrow = wrow0 + ky * WROW;
        #pragma unroll
        for (int ch = 0; ch < 7; ++ch) {
            v2f b;
            b.x = wrow[4 * ch];
            b.y = wrow[4 * ch + 1];
            v2f aL = *(const v2f*)(prow + 4 * ch);        // ds_load_b64, 8B aligned
            v2f aR = *(const v2f*)(prow + 4 * ch + 16);   // right half, +16 cols
            accL = __builtin_amdgcn_wmma_f32_16x16x4_f32(
                false, aL, false, b, (short)0, accL, false, false);
            accR = __builtin_amdgcn_wmma_f32_16x16x4_f32(
                false, aR, false, b, (short)0, accR, false, false);
        }
    }

    const float bc = bias[c];
    float* oc = out + ((size_t)(n * C_CH + c)) * (HW * HW);
    #pragma unroll
    for (int v = 0; v < 8; ++v) {
        int y = y0 + v + 8 * h;     // C/D layout: VGPR v -> row v (+8 upper half)
        int xL = x0 + mn;
        int xR = x0 + 16 + mn;
        if (y < HW) {
            if (xL < HW) oc[y * HW + xL] = accL[v] + bc;
            if (xR < HW) oc[y * HW + xR] = accR[v] + bc;
        }
    }
}

extern "C" void kernel_launch(void* const* d_in, const int* in_sizes, int n_in,
                              void* d_out, int out_size, void* d_ws, size_t ws_size,
                              hipStream_t stream) {
    (void)in_sizes; (void)n_in; (void)out_size; (void)ws_size;
    const float* x    = (const float*)d_in[0];
    const float* w_lk = (const float*)d_in[1];
    const float* w_b0 = (const float*)d_in[2];
    const float* w_b1 = (const float*)d_in[3];
    const float* w_b2 = (const float*)d_in[4];
    const float* w_b3 = (const float*)d_in[5];
    const float* w_b4 = (const float*)d_in[6];
    const float* w_b5 = (const float*)d_in[7];
    const float* g    = (const float*)d_in[8];
    const float* be   = (const float*)d_in[9];
    const float* mu   = (const float*)d_in[10];
    const float* va   = (const float*)d_in[11];

    float* Wm   = (float*)d_ws;            // 384*169 floats
    float* bias = Wm + C_CH * 169;         // 384 floats

    merge_weights_kernel<<<dim3(C_CH), dim3(192), 0, stream>>>(
        w_lk, w_b0, w_b1, w_b2, w_b3, w_b4, w_b5, g, be, mu, va, Wm, bias);

    dwconv13_wmma_kernel<<<dim3(8, C_CH, N_IMG), dim3(32), 0, stream>>>(
        x, Wm, bias, (float*)d_out);
}